// ProposalsLayer_87840671137982
// MI455X (gfx1250) — compile-verified
//
#include <hip/hip_runtime.h>
#include <hip/hip_bf16.h>
#include <math.h>
#include <stdint.h>

#ifndef __has_builtin
#define __has_builtin(x) 0
#endif

// ---------------- problem constants (from reference) ----------------
#define BATCH 4
#define NANCH 120000
#define NCLS 4
#define PRE_NMS 12000
#define POST_NMS 2000
#define IOU_THR 0.7f
#define NBINS 1024

typedef __attribute__((ext_vector_type(2))) float v2f;
typedef __attribute__((ext_vector_type(8))) float v8f;
typedef int v4i __attribute__((vector_size(16)));   // matches builtin param type

#define AS1 __attribute__((address_space(1)))
#define AS3 __attribute__((address_space(3)))

// ---------------- workspace layout (4-byte units) ----------------
#define HIST_OFF  0
#define CNTA_OFF  (BATCH * NBINS)
#define CNTT_OFF  (CNTA_OFF + BATCH)
#define THR_OFF   (CNTT_OFF + BATCH)
#define ABV_OFF   (THR_OFF + BATCH)
#define ZERO_N    (ABV_OFF + BATCH)          // region that must be zeroed each call
#define SCORE_OFF (ZERO_N)                   // B*NANCH floats
#define CS_OFF    (SCORE_OFF + BATCH * NANCH)         // candidate scores B*PRE_NMS
#define CB_OFF    (CS_OFF + BATCH * PRE_NMS)          // candidate boxes  B*PRE_NMS*4

// =====================================================================
// 0) zero the histogram + counters
// =====================================================================
__global__ void pl_init_kernel(int* __restrict__ wsi) {
    int i = blockIdx.x * blockDim.x + threadIdx.x;
    if (i < ZERO_N) wsi[i] = 0;
}

// =====================================================================
// 1) foreground max softmax score, denominator via v_wmma_f32_16x16x4_f32
//    One wave handles 16 anchors: lane L and lane L+16 share anchor L%16;
//    lane<16 holds classes {0,1}, lane>=16 holds classes {2,3}, which is
//    exactly the CDNA5 32-bit 16x4 A-matrix layout (VGPR0=K even, VGPR1=K odd).
//    D = A x ones(4x16): D[m][*] = sum_k exp(l[m][k]) = softmax denominator.
// =====================================================================
__global__ __launch_bounds__(256) void pl_score_kernel(
        const float* __restrict__ labels, float* __restrict__ scores) {
    const int lane  = threadIdx.x & 31;
    const int gwave = (blockIdx.x * blockDim.x + threadIdx.x) >> 5;
    const int a     = gwave * 16 + (lane & 15);      // anchor (flat over B*A)
    const int cpair = (lane >> 4) << 1;              // class pair base: 0 or 2

    const float2 l = *reinterpret_cast<const float2*>(labels + (size_t)a * NCLS + cpair);

    // per-anchor max over all 4 logits (pairwise + cross-half shuffle)
    float pm = fmaxf(l.x, l.y);
    float m  = fmaxf(pm, __shfl_xor(pm, 16, 32));

    float e0 = __expf(l.x - m);
    float e1 = __expf(l.y - m);

    v2f amat; amat[0] = e0; amat[1] = e1;            // 16x4 f32 A operand
    v2f bmat; bmat[0] = 1.0f; bmat[1] = 1.0f;        // 4x16 all-ones B operand
    v8f cacc = {};
    v8f d = __builtin_amdgcn_wmma_f32_16x16x4_f32(
        /*neg_a=*/false, amat, /*neg_b=*/false, bmat,
        /*c_mod=*/(short)0, cacc, /*reuse_a=*/false, /*reuse_b=*/false);

    // D layout: VGPR r holds row M=r (lanes 0-15) / M=r+8 (lanes 16-31).
    // Lane L wants denom of anchor L%16: select VGPR (L&7), fetch from the
    // lane half that owns rows (L%16)/8.
    const int r = lane & 7;
    float dsel = d[0];
    dsel = (r == 1) ? d[1] : dsel;
    dsel = (r == 2) ? d[2] : dsel;
    dsel = (r == 3) ? d[3] : dsel;
    dsel = (r == 4) ? d[4] : dsel;
    dsel = (r == 5) ? d[5] : dsel;
    dsel = (r == 6) ? d[6] : dsel;
    dsel = (r == 7) ? d[7] : dsel;
    const int srcl = (lane & 7) | ((lane & 8) << 1);
    float denom = __shfl(dsel, srcl, 32);

    // numerator: max over foreground classes 1..3 of exp(l - m)
    float cand = (lane < 16) ? e1 : fmaxf(e0, e1);
    float num  = fmaxf(cand, __shfl_xor(cand, 16, 32));

    if (lane < 16) scores[a] = num / denom;
}

// =====================================================================
// 2) per-batch 1024-bin histogram over positive-float bit patterns
// =====================================================================
__global__ void pl_hist_kernel(const float* __restrict__ scores, int* __restrict__ hist) {
    __shared__ int lh[NBINS];
    for (int i = threadIdx.x; i < NBINS; i += blockDim.x) lh[i] = 0;
    __syncthreads();
    const int b = blockIdx.y;
    const int stride = gridDim.x * blockDim.x;
    const float* sb = scores + (size_t)b * NANCH;
    for (int i = blockIdx.x * blockDim.x + threadIdx.x; i < NANCH; i += stride) {
        if (i + 4096 < NANCH) __builtin_prefetch(sb + i + 4096, 0, 1); // global_prefetch_b8
        unsigned bits = __float_as_uint(sb[i]);       // scores in (0,1): order-preserving
        atomicAdd(&lh[bits >> 22], 1);
    }
    __syncthreads();
    for (int i = threadIdx.x; i < NBINS; i += blockDim.x)
        if (lh[i]) atomicAdd(&hist[b * NBINS + i], lh[i]);
}

// =====================================================================
// 3) find threshold bin T: suffix count through T first reaches PRE_NMS
// =====================================================================
__global__ void pl_thresh_kernel(const int* __restrict__ hist,
                                 int* __restrict__ thr, int* __restrict__ abv) {
    if (threadIdx.x != 0) return;
    const int b = blockIdx.x;
    int acc = 0, T = 0, above = 0;
    for (int bin = NBINS - 1; bin >= 0; --bin) {
        int c = hist[b * NBINS + bin];
        if (acc + c >= PRE_NMS) { T = bin; above = acc; break; }
        acc += c;
        above = acc;
    }
    thr[b] = T;
    abv[b] = above;
}

// =====================================================================
// 4) compaction: all scores with bin>T, fill remainder from bin==T
// =====================================================================
__global__ void pl_compact_kernel(const float* __restrict__ scores,
                                  const float* __restrict__ boxes,
                                  const int* __restrict__ thr, const int* __restrict__ abv,
                                  int* __restrict__ cntA, int* __restrict__ cntT,
                                  float* __restrict__ cs, float* __restrict__ cb) {
    const int b = blockIdx.y;
    const int T = thr[b];
    const int above = abv[b];
    const int stride = gridDim.x * blockDim.x;
    const float* sb = scores + (size_t)b * NANCH;
    const float4* bb = reinterpret_cast<const float4*>(boxes) + (size_t)b * NANCH;
    for (int i = blockIdx.x * blockDim.x + threadIdx.x; i < NANCH; i += stride) {
        float s = sb[i];
        int bin = (int)(__float_as_uint(s) >> 22);
        int pos = -1;
        if (bin > T) {
            pos = atomicAdd(&cntA[b], 1);
        } else if (bin == T) {
            int p = above + atomicAdd(&cntT[b], 1);
            if (p < PRE_NMS) pos = p;
        }
        if (pos >= 0) {
            cs[(size_t)b * PRE_NMS + pos] = s;
            reinterpret_cast<float4*>(cb)[(size_t)b * PRE_NMS + pos] = bb[i];
        }
    }
}

// =====================================================================
// 5) greedy NMS, one workgroup per batch; entire candidate state (240 KB)
//    held in CDNA5's 320 KB LDS. Candidates loaded via async global->LDS
//    DMA when the gfx1250 builtins are available.
// =====================================================================
#if __has_builtin(__builtin_amdgcn_global_load_async_to_lds_b32)
#define HAVE_ASYNC32 1
#else
#define HAVE_ASYNC32 0
#endif
#if __has_builtin(__builtin_amdgcn_global_load_async_to_lds_b128)
#define HAVE_ASYNC128 1
#else
#define HAVE_ASYNC128 0
#endif

__device__ __forceinline__ void pl_wait_async() {
#if __has_builtin(__builtin_amdgcn_s_wait_asynccnt)
    __builtin_amdgcn_s_wait_asynccnt(0);
#else
    asm volatile("s_wait_asynccnt 0" ::: "memory");
#endif
}

__global__ __launch_bounds__(1024) void pl_nms_kernel(
        const float* __restrict__ cs, const float* __restrict__ cb,
        float* __restrict__ out) {
    extern __shared__ char smem[];
    float*  ss = reinterpret_cast<float*>(smem);          // [PRE_NMS] scores
    float4* sb = reinterpret_cast<float4*>(ss + PRE_NMS); // [PRE_NMS] boxes
    float*  rs = reinterpret_cast<float*>(sb + PRE_NMS);  // [32] wave best score
    int*    ri = reinterpret_cast<int*>(rs + 32);         // [32] wave best idx
    float*  bc = reinterpret_cast<float*>(ri + 32);       // [8] broadcast box+flag

    const int tid  = threadIdx.x;
    const int b    = blockIdx.x;
    const int lane = tid & 31;
    const int wid  = tid >> 5;

    const float*  gs = cs + (size_t)b * PRE_NMS;
    const float4* gb = reinterpret_cast<const float4*>(cb) + (size_t)b * PRE_NMS;

    // ---- gather candidates into LDS ----
    for (int i = tid; i < PRE_NMS; i += 1024) {
#if HAVE_ASYNC32
        __builtin_amdgcn_global_load_async_to_lds_b32(
            (AS1 int*)(gs + i), (AS3 int*)(ss + i), 0, 0);
#else
        ss[i] = gs[i];
#endif
#if HAVE_ASYNC128
        __builtin_amdgcn_global_load_async_to_lds_b128(
            (AS1 v4i*)(gb + i), (AS3 v4i*)(sb + i), 0, 0);
#else
        sb[i] = gb[i];
#endif
    }
#if HAVE_ASYNC32 || HAVE_ASYNC128
    pl_wait_async();
#endif
    __syncthreads();

    // ---- sequential greedy loop ----
    for (int it = 0; it < POST_NMS; ++it) {
        // thread-local argmax over strided candidates
        float best = -1.0f;
        int bi = 0;
        for (int i = tid; i < PRE_NMS; i += 1024) {
            float s = ss[i];
            if (s > best) { best = s; bi = i; }
        }
        // wave32 reduction
        for (int off = 16; off > 0; off >>= 1) {
            float os = __shfl_down(best, off, 32);
            int   oi = __shfl_down(bi, off, 32);
            if (os > best) { best = os; bi = oi; }
        }
        if (lane == 0) { rs[wid] = best; ri[wid] = bi; }
        __syncthreads();
        // cross-wave reduction by wave 0
        if (wid == 0) {
            best = rs[lane];
            bi   = ri[lane];
            for (int off = 16; off > 0; off >>= 1) {
                float os = __shfl_down(best, off, 32);
                int   oi = __shfl_down(bi, off, 32);
                if (os > best) { best = os; bi = oi; }
            }
            if (lane == 0) {
                if (best > 0.0f) {
                    float4 box = sb[bi];
                    ss[bi] = -1.0f;
                    bc[0] = box.x; bc[1] = box.y; bc[2] = box.z; bc[3] = box.w;
                    bc[4] = 1.0f;
                    float4 o;
                    o.x = fminf(fmaxf(box.x, 0.0f), 1.0f);
                    o.y = fminf(fmaxf(box.y, 0.0f), 1.0f);
                    o.z = fminf(fmaxf(box.z, 0.0f), 1.0f);
                    o.w = fminf(fmaxf(box.w, 0.0f), 1.0f);
                    reinterpret_cast<float4*>(out)[(size_t)b * POST_NMS + it] = o;
                } else {
                    bc[4] = 0.0f;
                }
            }
        }
        __syncthreads();

        if (bc[4] == 0.0f) {
            // no valid boxes remain: zero-fill the rest (reference zero-pads)
            for (int j = it * 4 + tid; j < POST_NMS * 4; j += 1024)
                out[(size_t)b * POST_NMS * 4 + j] = 0.0f;
            break;
        }

        // suppress by IoU against the selected box
        const float y1 = bc[0], x1 = bc[1], y2 = bc[2], x2 = bc[3];
        const float areaA = (y2 - y1) * (x2 - x1);
        for (int i = tid; i < PRE_NMS; i += 1024) {
            float s = ss[i];
            if (s > 0.0f) {
                float4 bx = sb[i];
                float iy1 = fmaxf(y1, bx.x);
                float ix1 = fmaxf(x1, bx.y);
                float iy2 = fminf(y2, bx.z);
                float ix2 = fminf(x2, bx.w);
                float ih = fmaxf(iy2 - iy1, 0.0f);
                float iw = fmaxf(ix2 - ix1, 0.0f);
                float inter = ih * iw;
                float areaB = (bx.z - bx.x) * (bx.w - bx.y);
                float uni = fmaxf(areaA + areaB - inter, 1e-8f);
                if (inter > IOU_THR * uni) ss[i] = -1.0f;
            }
        }
        __syncthreads();
    }
}

// =====================================================================
// launch
// =====================================================================
extern "C" void kernel_launch(void* const* d_in, const int* in_sizes, int n_in,
                              void* d_out, int out_size, void* d_ws, size_t ws_size,
                              hipStream_t stream) {
    const float* boxes  = (const float*)d_in[0];   // [B, A, 4]
    const float* labels = (const float*)d_in[1];   // [B, A, C]
    float* out = (float*)d_out;                    // [B, POST_NMS, 4]
    int*   wsi = (int*)d_ws;
    float* wsf = (float*)d_ws;

    // 0) zero histogram + counters
    pl_init_kernel<<<(ZERO_N + 255) / 256, 256, 0, stream>>>(wsi);

    // 1) scores: 8 waves/block * 16 anchors/wave = 128 anchors/block (exact)
    pl_score_kernel<<<(BATCH * NANCH) / 128, 256, 0, stream>>>(labels, wsf + SCORE_OFF);

    // 2) histogram
    dim3 hg(32, BATCH);
    pl_hist_kernel<<<hg, 256, 0, stream>>>(wsf + SCORE_OFF, wsi + HIST_OFF);

    // 3) threshold bin
    pl_thresh_kernel<<<BATCH, 32, 0, stream>>>(wsi + HIST_OFF, wsi + THR_OFF, wsi + ABV_OFF);

    // 4) compact to exactly PRE_NMS candidates per batch
    dim3 cg(64, BATCH);
    pl_compact_kernel<<<cg, 256, 0, stream>>>(wsf + SCORE_OFF, boxes,
                                              wsi + THR_OFF, wsi + ABV_OFF,
                                              wsi + CNTA_OFF, wsi + CNTT_OFF,
                                              wsf + CS_OFF, wsf + CB_OFF);

    // 5) NMS: 240 KB of dynamic LDS per workgroup (CDNA5 WGP has 320 KB)
    size_t lds_bytes = (size_t)PRE_NMS * sizeof(float)      // scores
                     + (size_t)PRE_NMS * sizeof(float4)     // boxes
                     + 32 * sizeof(float) + 32 * sizeof(int) + 8 * sizeof(float);
    pl_nms_kernel<<<BATCH, 1024, lds_bytes, stream>>>(wsf + CS_OFF, wsf + CB_OFF, out);
}